// EmbeddingCFFModel_79491254715037
// MI455X (gfx1250) — compile-verified
//
#include <hip/hip_runtime.h>
#include <math.h>

// ---------------------------------------------------------------------------
// Stage 1: per-point coefficients (P = 4096). All phi-independent algebra of
// the BKM10 surrogate hoisted here. 16-float (64B) record per point ->
// 256 KB in d_ws, L2-resident (192 MB) for stage 2.
// ---------------------------------------------------------------------------

#define PP_STRIDE 16  // floats per point record (64B aligned)

__global__ void cff_point_kernel(const float* __restrict__ theta,
                                 const float* __restrict__ kin,
                                 float* __restrict__ coef, int P)
{
    int p = blockIdx.x * blockDim.x + threadIdx.x;
    if (p >= P) return;

    // CFF_SCALES = [1,1,1,150, 1,1,1,150]
    const float reH  = theta[p * 8 + 0];
    const float reE  = theta[p * 8 + 1];
    const float reHt = theta[p * 8 + 2];
    const float reEt = theta[p * 8 + 3] * 150.0f;
    const float imH  = theta[p * 8 + 4];
    const float imE  = theta[p * 8 + 5];
    const float imHt = theta[p * 8 + 6];
    const float imEt = theta[p * 8 + 7] * 150.0f;

    const float t  = kin[p * 3 + 0];
    const float xb = kin[p * 3 + 1];
    const float q2 = kin[p * 3 + 2];

    const float MU_P   = 2.79285f;
    const float M2f    = (float)(0.938272 * 0.938272);
    const float TWO_MK = (float)(2.0 * 0.938272 * 5.75);
    const float FOURM2 = (float)(4.0 * 0.938272 * 0.938272);
    const float DIPOLE = 0.710649f;

    const float y    = q2 / (TWO_MK * xb);
    const float eps2 = 4.0f * xb * xb * M2f / q2;
    const float xi   = xb * (1.0f + t / (2.0f * q2)) / (2.0f - xb + xb * t / q2);
    const float tau  = -t / FOURM2;
    const float gdd  = 1.0f - t / DIPOLE;
    const float gd   = 1.0f / (gdd * gdd);
    const float F1   = gd * (1.0f + tau * MU_P) / (1.0f + tau);
    const float F2   = gd * (MU_P - 1.0f) / (1.0f + tau);

    const float K2 = fmaxf(-t / q2 * (1.0f - xb) * (1.0f - y - 0.25f * y * y * eps2), 1e-8f);
    const float K  = sqrtf(K2);
    const float tw3 = 2.0f * xi / (1.0f + xi);   // using_ww = True

    const float xiF12 = xi * (F1 + F2);
    const float CI_re = F1 * reH + xiF12 * reHt - tau * F2 * reE;
    const float CI_im = F1 * imH + xiF12 * imHt - tau * F2 * imE;
    const float CT_re = F1 * reHt + xiF12 * reH - tau * xi * F2 * reEt;
    const float CT_im = F1 * imHt + xiF12 * imH - tau * xi * F2 * imEt;

    const float c_dvcs =
        4.0f * (1.0f - xb) / ((2.0f - xb) * (2.0f - xb)) *
            (reH * reH + imH * imH + xi * xi * (reHt * reHt + imHt * imHt))
        - tau * (reE * reE + imE * imE + xi * xi * (reEt * reEt + imEt * imEt));

    const float onep = 1.0f + eps2;
    const float pre  = 1.0f / (xb * y * y * onep * onep * fmaxf(-t, 1e-6f));
    const float y_fac = 2.0f - 2.0f * y + y * y;

    const float F12  = F1 + F2;
    const float bh_n = pre * (8.0f * K2 * (F1 * F1 + tau * F2 * F2)
                              + 2.0f * tau * y_fac * F12 * F12);
    const float dvcs = c_dvcs * y_fac / (y * y * q2);

    const float c_bca  = pre * 8.0f * K * y_fac * CI_re;
    const float c_bsa  = pre * 8.0f * K * y * (2.0f - y) * CI_im;
    const float c_tsa  = pre * 8.0f * K * y * (2.0f - y) * CT_im;
    const float c_dsa1 = pre * (2.0f - y) * 8.0f * K * y * CT_re;
    const float c_dsa0 = pre * (2.0f - y) * 4.0f * y * tau * F12 * (F1 + tau * F2);

    float4* rec = (float4*)(coef + (size_t)p * PP_STRIDE);
    rec[0] = make_float4(K2, 2.0f * K, tw3 * K, bh_n);
    rec[1] = make_float4(dvcs, c_bca, c_bsa, c_tsa);
    rec[2] = make_float4(c_dsa1, c_dsa0, 0.0f, 0.0f);
    rec[3] = make_float4(0.0f, 0.0f, 0.0f, 0.0f);
}

// ---------------------------------------------------------------------------
// Stage 2: per-sample kernel, 4 samples per thread.
//   - phi staged via gfx1250 async global->LDS b128 (ASYNCcnt path),
//     read back through an explicit AS(3) volatile pointer (ds_load).
//   - point_id as int4, outputs as 5 aligned float4 stores per thread.
//   - coefficient record loaded once per 4-sample group when the group is
//     point-uniform (always true for the 360-phi layout), per-sample otherwise.
// Builtin prototype (from hipcc diagnostics across rounds):
//   b128: (as1 v4i* gsrc, as3 v4i* ldst, imm offset, imm cpol)
// ---------------------------------------------------------------------------

#define HAS_ASYNC_LDS \
    (__has_builtin(__builtin_amdgcn_global_load_async_to_lds_b128) && \
     __has_builtin(__builtin_amdgcn_s_wait_asynccnt))

typedef int v4i __attribute__((ext_vector_type(4)));
typedef __attribute__((address_space(1))) v4i* gas_v4p;
typedef __attribute__((address_space(3))) v4i* las_v4p;
typedef __attribute__((address_space(3))) volatile float* lvf_p;

__device__ __forceinline__ void eval_sample(float phi,
                                            float K2, float twoK, float tw3K,
                                            float bh_n, float dvcs,
                                            float c_bca, float c_bsa, float c_tsa,
                                            float c_dsa1, float c_dsa0,
                                            float* __restrict__ r)
{
    float sphi, cphi;
    __sincosf(phi, &sphi, &cphi);

    const float a    = 1.0f + K2;
    const float u    = twoK * cphi;
    const float P1P2 = (a + u) * (a - u);
    const float rp   = 1.0f / P1P2;

    const float i_unpol = c_bca * (cphi + tw3K * (2.0f * cphi * cphi - 1.0f)) * rp;
    const float xs      = bh_n * rp + dvcs + i_unpol;
    const float xs_safe = (fabsf(xs) > 1e-8f) ? xs : 1e-8f;
    const float inv_xs  = 1.0f / xs_safe;

    const float srp = sphi * rp * inv_xs;
    const float crp = cphi * rp * inv_xs;
    r[0] = xs;
    r[1] = c_bsa * srp;
    r[2] = c_bca * crp;
    r[3] = c_tsa * srp;
    r[4] = (c_dsa1 * cphi + c_dsa0) * rp * inv_xs;
}

__global__ void cff_sample_kernel(const int* __restrict__ point_id,
                                  const float* __restrict__ phi_g,
                                  const float* __restrict__ coef,
                                  float* __restrict__ out, int N, int G)
{
    __shared__ float phi_lds[256 * 4];

    const int g = blockIdx.x * 256 + threadIdx.x;
    if (g >= G) return;
    const int s0 = g * 4;

    const bool full = (s0 + 3) < N;

    float ph[4];
    int   pid[4];

    if (full) {
#if HAS_ASYNC_LDS
        // gfx1250 async global->LDS staging of this thread's 4 phi values.
        __builtin_amdgcn_global_load_async_to_lds_b128(
            (gas_v4p)(phi_g + s0), (las_v4p)(&phi_lds[threadIdx.x * 4]), 0, 0);
#endif
        const int4 p4 = *(const int4*)(point_id + s0);
        pid[0] = p4.x; pid[1] = p4.y; pid[2] = p4.z; pid[3] = p4.w;
#if HAS_ASYNC_LDS
        __builtin_amdgcn_s_wait_asynccnt(0);
        lvf_p lp = (lvf_p)(&phi_lds[threadIdx.x * 4]);
        ph[0] = lp[0]; ph[1] = lp[1]; ph[2] = lp[2]; ph[3] = lp[3];
#else
        const float4 f4 = *(const float4*)(phi_g + s0);
        ph[0] = f4.x; ph[1] = f4.y; ph[2] = f4.z; ph[3] = f4.w;
#endif
    } else {
        for (int i = 0; i < 4; ++i) {
            const int s = s0 + i;
            const int ss = (s < N) ? s : (N - 1);
            ph[i]  = phi_g[ss];
            pid[i] = point_id[ss];
        }
    }

    float r[20];

    if (pid[0] == pid[3] && pid[0] == pid[1] && pid[0] == pid[2]) {
        // point-uniform group (the common case for the 360-phi layout)
        const float4* rec = (const float4*)(coef + (size_t)pid[0] * PP_STRIDE);
        const float4 c0 = rec[0];
        const float4 c1 = rec[1];
        const float4 c2 = rec[2];
        #pragma unroll
        for (int i = 0; i < 4; ++i)
            eval_sample(ph[i], c0.x, c0.y, c0.z, c0.w, c1.x, c1.y, c1.z, c1.w,
                        c2.x, c2.y, &r[i * 5]);
    } else {
        #pragma unroll
        for (int i = 0; i < 4; ++i) {
            const float4* rec = (const float4*)(coef + (size_t)pid[i] * PP_STRIDE);
            const float4 c0 = rec[0];
            const float4 c1 = rec[1];
            const float4 c2 = rec[2];
            eval_sample(ph[i], c0.x, c0.y, c0.z, c0.w, c1.x, c1.y, c1.z, c1.w,
                        c2.x, c2.y, &r[i * 5]);
        }
    }

    if (full) {
        float4* o = (float4*)(out + (size_t)s0 * 5);   // 80B base, 16B aligned
        o[0] = make_float4(r[0],  r[1],  r[2],  r[3]);
        o[1] = make_float4(r[4],  r[5],  r[6],  r[7]);
        o[2] = make_float4(r[8],  r[9],  r[10], r[11]);
        o[3] = make_float4(r[12], r[13], r[14], r[15]);
        o[4] = make_float4(r[16], r[17], r[18], r[19]);
    } else {
        for (int i = 0; i < 4; ++i) {
            const int s = s0 + i;
            if (s < N) {
                float* o = out + (size_t)s * 5;
                o[0] = r[i * 5 + 0];
                o[1] = r[i * 5 + 1];
                o[2] = r[i * 5 + 2];
                o[3] = r[i * 5 + 3];
                o[4] = r[i * 5 + 4];
            }
        }
    }
}

// ---------------------------------------------------------------------------
// Launch. Inputs in setup_inputs() order:
//   d_in[0] theta (P*8 f32), d_in[1] unique_kin (P*3 f32),
//   d_in[2] point_id (N i32), d_in[3] phi (N f32)
// d_out: N*5 f32; d_ws: >= P*PP_STRIDE*4 bytes (256 KB at P=4096)
// ---------------------------------------------------------------------------
extern "C" void kernel_launch(void* const* d_in, const int* in_sizes, int n_in,
                              void* d_out, int out_size, void* d_ws, size_t ws_size,
                              hipStream_t stream)
{
    const float* theta = (const float*)d_in[0];
    const float* kin   = (const float*)d_in[1];
    const int*   pid   = (const int*)d_in[2];
    const float* phi   = (const float*)d_in[3];
    float*       out   = (float*)d_out;
    float*       coef  = (float*)d_ws;

    const int P = in_sizes[0] / 8;
    const int N = in_sizes[2];
    const int G = (N + 3) / 4;

    cff_point_kernel<<<(P + 255) / 256, 256, 0, stream>>>(theta, kin, coef, P);
    cff_sample_kernel<<<(G + 255) / 256, 256, 0, stream>>>(pid, phi, coef, out, N, G);
}